// CVSSTL_26027501814276
// MI455X (gfx1250) — compile-verified
//
#include <hip/hip_runtime.h>
#include <math.h>

// Problem constants (from reference)
#define BB 8
#define NN 4096
#define CC 256
#define HH 8
#define WSZ 64
#define HD 32
#define TT (BB * NN)          // 32768 tokens
#define MLP 1024
#define QK_SCALE 0.17677669529663687f  // 32^-0.5

typedef __attribute__((ext_vector_type(2))) float v2f;
typedef __attribute__((ext_vector_type(8))) float v8f;

// ---------------------------------------------------------------------------
// Weight packing: complex (O, Kc) weight -> real (2Kc, 2O):
//   cols [0,O): real out  = [Wr ; -Wi],  cols [O,2O): imag out = [Wi ; Wr]
// so  [Ar | Ai] (T x 2Kc)  @  Wp (2Kc x 2O)  =  [Or | Oi]  (T x 2O)
// ---------------------------------------------------------------------------
__global__ void pack_weight(const float* __restrict__ w, float* __restrict__ Wp,
                            int O, int Kc) {
  const int twoO = 2 * O;
  const size_t total = (size_t)(2 * Kc) * twoO;
  for (size_t idx = (size_t)blockIdx.x * blockDim.x + threadIdx.x; idx < total;
       idx += (size_t)gridDim.x * blockDim.x) {
    const int n = (int)(idx % twoO);
    const int k = (int)(idx / twoO);
    const int o = (n < O) ? n : n - O;
    float val;
    if (n < O) {  // real output column
      if (k < Kc) val =  w[2 * ((size_t)o * Kc + k)];            // Wr
      else        val = -w[2 * ((size_t)o * Kc + (k - Kc)) + 1]; // -Wi
    } else {      // imag output column
      if (k < Kc) val =  w[2 * ((size_t)o * Kc + k) + 1];        // Wi
      else        val =  w[2 * ((size_t)o * Kc + (k - Kc))];     // Wr
    }
    Wp[idx] = val;
  }
}

__global__ void pack_bias(const float* __restrict__ b, float* __restrict__ bp,
                          int O) {
  const int o = blockIdx.x * blockDim.x + threadIdx.x;
  if (o < O) {
    bp[o]     = b[2 * o];      // real
    bp[O + o] = b[2 * o + 1];  // imag
  }
}

// ---------------------------------------------------------------------------
// Complex LayerNorm: separate LN over C=256 channels of real and imag parts,
// then complex gamma/beta. One wave (32 lanes) per token, 8 channels/lane.
// Output packed: out[t*512 + c] = real, out[t*512 + 256 + c] = imag.
// ---------------------------------------------------------------------------
__global__ __launch_bounds__(256) void cln_kernel(
    const float* __restrict__ in_r, const float* __restrict__ in_i, int stride,
    const float* __restrict__ g, const float* __restrict__ bb,
    float* __restrict__ out) {
  const int wave = threadIdx.x >> 5;
  const int lane = threadIdx.x & 31;
  const size_t t = (size_t)blockIdx.x * 8 + wave;
  const float* pr = in_r + t * stride;
  const float* pi = in_i + t * stride;

  float xr[8], xi[8];
  float sr = 0.f, srr = 0.f, si = 0.f, sii = 0.f;
#pragma unroll
  for (int j = 0; j < 8; ++j) {
    const int c = lane + j * 32;
    xr[j] = pr[c];
    xi[j] = pi[c];
    sr += xr[j]; srr += xr[j] * xr[j];
    si += xi[j]; sii += xi[j] * xi[j];
  }
#pragma unroll
  for (int m = 16; m >= 1; m >>= 1) {
    sr  += __shfl_xor(sr, m, 32);
    srr += __shfl_xor(srr, m, 32);
    si  += __shfl_xor(si, m, 32);
    sii += __shfl_xor(sii, m, 32);
  }
  const float mr = sr * (1.f / 256.f);
  const float mi = si * (1.f / 256.f);
  const float vr = srr * (1.f / 256.f) - mr * mr;
  const float vi = sii * (1.f / 256.f) - mi * mi;
  const float invr = rsqrtf(vr + 1e-5f);
  const float invi = rsqrtf(vi + 1e-5f);

#pragma unroll
  for (int j = 0; j < 8; ++j) {
    const int c = lane + j * 32;
    const float nr = (xr[j] - mr) * invr;
    const float ni = (xi[j] - mi) * invi;
    const float gr = g[2 * c], gi = g[2 * c + 1];
    const float br = bb[2 * c], bi = bb[2 * c + 1];
    out[t * 512 + c]       = nr * gr - ni * gi + br;
    out[t * 512 + 256 + c] = nr * gi + ni * gr + bi;
  }
}

// ---------------------------------------------------------------------------
// Generic f32 WMMA GEMM:  C(MxN) = A(MxK) @ W(KxN) + bias, fused epilogues.
// Workgroup: 256 threads (8 waves), 128x128 tile. Each wave: 64x32 (4x2
// accumulators of 16x16) via v_wmma_f32_16x16x4_f32.
//
// K is staged 32-deep in double-buffered LDS, filled with the CDNA5 async
// path (GLOBAL_LOAD_ASYNC_TO_LDS_B128, ASYNCcnt): the next tile streams
// HBM->LDS (bypassing VGPRs) while the current tile feeds the WMMAs.
// Per iteration: s_wait_asynccnt 0 (own fills landed) + s_barrier (everyone
// finished previous compute) -> safe to read cur buffer / refill alt buffer.
//
// epi: 0 = bias only (qkv)
//      1 = bias + complex PReLU (fc1): slope per real/imag half
//      2 = bias + residual from two planes resA/resB (proj: + shortcut x)
//      3 = bias + packed residual resA, write split real/imag planes (fc2)
// ---------------------------------------------------------------------------
__global__ __launch_bounds__(256) void gemm_wmma_f32(
    const float* __restrict__ A, const float* __restrict__ W,
    const float* __restrict__ bias, float* __restrict__ Cout,
    int M, int K, int N, int epi, int halfN,
    const float* __restrict__ resA, const float* __restrict__ resB,
    const float* __restrict__ slopes, float* __restrict__ out2) {
  __shared__ float As[2][128][32];   // 2 x 16 KB
  __shared__ float Bs[2][32][128];   // 2 x 16 KB

  const int tid  = threadIdx.x;
  const int lane = tid & 31;
  const int wave = tid >> 5;
  const int wm = wave >> 2;   // 0..1  (M sub-blocks of 64)
  const int wn = wave & 3;    // 0..3  (N sub-blocks of 32)
  const int hi = lane >> 4;   // half-wave select
  const int lx = lane & 15;
  const int m0 = blockIdx.y * 128;
  const int n0 = blockIdx.x * 128;

  // Issue async HBM->LDS fills for one 128x32 A tile + 32x128 B tile.
  // Each wave: 4 + 4 B128 ops (512 B per wave-op); ASYNCcnt is per-wave.
  auto issue_tile = [&](int buf, int kt) {
#pragma unroll
    for (int i = 0; i < 4; ++i) {
      const int l = wave * 128 + i * 32 + lane;  // 0..1023 float4 slots
      {
        const int r = l >> 3, cb = (l & 7) << 2;
        const unsigned int ldsA = (unsigned int)(size_t)&As[buf][r][cb];
        const unsigned long long gA =
            (unsigned long long)(size_t)&A[(size_t)(m0 + r) * K + kt + cb];
        asm volatile("global_load_async_to_lds_b128 %0, %1, off"
                     :: "v"(ldsA), "v"(gA) : "memory");
      }
      {
        const int r = l >> 5, cb = (l & 31) << 2;
        const unsigned int ldsB = (unsigned int)(size_t)&Bs[buf][r][cb];
        const unsigned long long gB =
            (unsigned long long)(size_t)&W[(size_t)(kt + r) * N + n0 + cb];
        asm volatile("global_load_async_to_lds_b128 %0, %1, off"
                     :: "v"(ldsB), "v"(gB) : "memory");
      }
    }
  };

  const v8f zero = {0.f, 0.f, 0.f, 0.f, 0.f, 0.f, 0.f, 0.f};
  v8f acc[4][2];
#pragma unroll
  for (int mi = 0; mi < 4; ++mi)
#pragma unroll
    for (int ni = 0; ni < 2; ++ni) acc[mi][ni] = zero;

  issue_tile(0, 0);
  int buf = 0;
  for (int kt = 0; kt < K; kt += 32) {
    asm volatile("s_wait_asynccnt 0x0" ::: "memory");
    __syncthreads();
    if (kt + 32 < K) issue_tile(buf ^ 1, kt + 32);

#pragma unroll
    for (int k4 = 0; k4 < 8; ++k4) {
      const int kk = k4 * 4;
      v2f a[4], b[2];
      // A fragment 16x4: lanes 0-15 rows M, VGPR0 = K(+2*hi), VGPR1 = K+1(+2*hi)
#pragma unroll
      for (int mi = 0; mi < 4; ++mi) {
        const int r = wm * 64 + mi * 16 + lx;
        a[mi].x = As[buf][r][kk + 2 * hi];
        a[mi].y = As[buf][r][kk + 2 * hi + 1];
      }
      // B fragment 4x16: rows K striped across VGPR/half-wave, cols across lanes
#pragma unroll
      for (int ni = 0; ni < 2; ++ni) {
        const int c = wn * 32 + ni * 16 + lx;
        b[ni].x = Bs[buf][kk + 2 * hi][c];
        b[ni].y = Bs[buf][kk + 2 * hi + 1][c];
      }
#pragma unroll
      for (int mi = 0; mi < 4; ++mi)
#pragma unroll
        for (int ni = 0; ni < 2; ++ni)
          acc[mi][ni] = __builtin_amdgcn_wmma_f32_16x16x4_f32(
              false, a[mi], false, b[ni], (short)0, acc[mi][ni], false, false);
    }
    buf ^= 1;
  }

  // Epilogue: C/D layout: lane lx = col, VGPR v -> row v + 8*hi within tile.
#pragma unroll
  for (int mi = 0; mi < 4; ++mi)
#pragma unroll
    for (int ni = 0; ni < 2; ++ni)
#pragma unroll
      for (int v = 0; v < 8; ++v) {
        const int row = m0 + wm * 64 + mi * 16 + v + 8 * hi;
        const int col = n0 + wn * 32 + ni * 16 + lx;
        const size_t t = (size_t)row;
        float val = acc[mi][ni][v] + bias[col];
        if (epi == 0) {
          Cout[t * N + col] = val;
        } else if (epi == 1) {
          const float s = (col < halfN) ? slopes[0] : slopes[1];
          Cout[t * N + col] = (val >= 0.f) ? val : s * val;
        } else if (epi == 2) {
          val += (col < halfN) ? resA[t * halfN + col]
                               : resB[t * halfN + (col - halfN)];
          Cout[t * N + col] = val;
        } else {  // epi == 3: packed residual, split real/imag planes
          val += resA[t * N + col];
          if (col < halfN) Cout[t * (size_t)halfN + col] = val;
          else             out2[t * (size_t)halfN + (col - halfN)] = val;
        }
      }
}

// ---------------------------------------------------------------------------
// Windowed complex attention with phase softmax.
// One block per (window, head). 512 windows x 8 heads = 4096 blocks.
// LDS: q/k (later v) in buf (32 KB) + Sr/Si (32 KB) = 64 KB.
// Only ~4% of total FLOPs -> VALU is fine here.
// qkv layout per token (1536 floats): [qr kr vr | qi ki vi], 256 each,
// head h occupies channels [h*32, h*32+32).
// ---------------------------------------------------------------------------
__global__ __launch_bounds__(256) void attn_kernel(
    const float* __restrict__ qkv, const float* __restrict__ rpb,
    float* __restrict__ aout) {
  __shared__ float buf[4][64][32];  // qr,qi,kr,ki ; later buf[0]=vr, buf[1]=vi
  __shared__ float Sr[64][64];
  __shared__ float Si[64][64];

  const int win = blockIdx.x >> 3;
  const int h   = blockIdx.x & 7;
  const int tid = threadIdx.x;
  const int r   = tid >> 2;          // row 0..63
  const int db  = (tid & 3) * 8;     // 8-wide channel slab
  const size_t tokBase = (size_t)(win * 64 + r) * 1536;
  const int hc = h * 32;

  // Load q (pre-scaled) and k
#pragma unroll
  for (int j = 0; j < 8; ++j) {
    const int d = db + j;
    buf[0][r][d] = qkv[tokBase + hc + d] * QK_SCALE;          // qr
    buf[1][r][d] = qkv[tokBase + 768 + hc + d] * QK_SCALE;    // qi
    buf[2][r][d] = qkv[tokBase + 256 + hc + d];               // kr
    buf[3][r][d] = qkv[tokBase + 768 + 256 + hc + d];         // ki
  }
  __syncthreads();

  // Phase 1: S = q k^T (complex) + relative position bias
  const int q  = tid >> 2;
  const int kb = (tid & 3) * 16;
  for (int k = kb; k < kb + 16; ++k) {
    float sr = 0.f, si = 0.f;
#pragma unroll
    for (int d = 0; d < 32; ++d) {
      const float qr_ = buf[0][q][d], qi_ = buf[1][q][d];
      const float kr_ = buf[2][k][d], ki_ = buf[3][k][d];
      sr += qr_ * kr_ - qi_ * ki_;
      si += qr_ * ki_ + qi_ * kr_;
    }
    const int idx = q - k + 63;
    sr += rpb[2 * (idx * 8 + h)];
    si += rpb[2 * (idx * 8 + h) + 1];
    Sr[q][k] = sr;
    Si[q][k] = si;
  }
  __syncthreads();

  // Load v into buf[0]/buf[1] (q/k dead); in parallel threads<64 do softmax
#pragma unroll
  for (int j = 0; j < 8; ++j) {
    const int d = db + j;
    buf[0][r][d] = qkv[tokBase + 512 + hc + d];
    buf[1][r][d] = qkv[tokBase + 768 + 512 + hc + d];
  }
  if (tid < 64) {
    const int row = tid;
    float mx = -1e30f;
    for (int k = 0; k < 64; ++k) {
      const float a = Sr[row][k], b = Si[row][k];
      mx = fmaxf(mx, sqrtf(a * a + b * b));
    }
    float sum = 0.f;
    for (int k = 0; k < 64; ++k) {
      const float a = Sr[row][k], b = Si[row][k];
      sum += expf(sqrtf(a * a + b * b) - mx);
    }
    const float inv = 1.f / sum;
    for (int k = 0; k < 64; ++k) {
      const float a = Sr[row][k], b = Si[row][k];
      const float m = sqrtf(a * a + b * b);
      const float sc = expf(m - mx) * inv / fmaxf(m, 1e-9f);
      Sr[row][k] = a * sc;
      Si[row][k] = b * sc;
    }
  }
  __syncthreads();

  // Phase 3: out = S @ v (complex); write packed [Or | Oi] per token
  for (int j = 0; j < 8; ++j) {
    const int d = db + j;
    float orv = 0.f, oiv = 0.f;
    for (int k = 0; k < 64; ++k) {
      const float sr_ = Sr[r][k], si_ = Si[r][k];
      const float vr_ = buf[0][k][d], vi_ = buf[1][k][d];
      orv += sr_ * vr_ - si_ * vi_;
      oiv += sr_ * vi_ + si_ * vr_;
    }
    const size_t tok = (size_t)(win * 64 + r);
    aout[tok * 512 + hc + d]       = orv;
    aout[tok * 512 + 256 + hc + d] = oiv;
  }
}

// ---------------------------------------------------------------------------
// Launch
// ---------------------------------------------------------------------------
extern "C" void kernel_launch(void* const* d_in, const int* in_sizes, int n_in,
                              void* d_out, int out_size, void* d_ws,
                              size_t ws_size, hipStream_t stream) {
  const float* x_r    = (const float*)d_in[0];
  const float* x_i    = (const float*)d_in[1];
  const float* n1g    = (const float*)d_in[2];
  const float* n1b    = (const float*)d_in[3];
  const float* qkv_w  = (const float*)d_in[4];
  const float* qkv_b  = (const float*)d_in[5];
  const float* rpb    = (const float*)d_in[6];
  const float* proj_w = (const float*)d_in[7];
  const float* proj_b = (const float*)d_in[8];
  const float* n2g    = (const float*)d_in[9];
  const float* n2b    = (const float*)d_in[10];
  const float* fc1_w  = (const float*)d_in[11];
  const float* fc1_b  = (const float*)d_in[12];
  const float* fc2_w  = (const float*)d_in[13];
  const float* fc2_b  = (const float*)d_in[14];
  const float* slopes = (const float*)d_in[15];

  float* ws = (float*)d_ws;
  size_t off = 0;
  float* Wqkv = ws + off; off += (size_t)512 * 1536;
  float* bqkv = ws + off; off += 1536;
  float* Wprj = ws + off; off += (size_t)512 * 512;
  float* bprj = ws + off; off += 512;
  float* Wfc1 = ws + off; off += (size_t)512 * 2048;
  float* bfc1 = ws + off; off += 2048;
  float* Wfc2 = ws + off; off += (size_t)2048 * 512;
  float* bfc2 = ws + off; off += 512;
  float* xn   = ws + off; off += (size_t)TT * 512;   // LN output (reused for LN2)
  float* qkv  = ws + off; off += (size_t)TT * 1536;  // qkv activations
  float* aout = ws + off; off += (size_t)TT * 512;   // attention output
  float* x1   = ws + off; off += (size_t)TT * 512;   // post-attention residual
  float* h1   = ws + off; off += (size_t)TT * 2048;  // fc1 output

  // Pack complex weights/biases into real-GEMM form
  pack_weight<<<1024, 256, 0, stream>>>(qkv_w, Wqkv, 768, 256);
  pack_bias<<<3, 256, 0, stream>>>(qkv_b, bqkv, 768);
  pack_weight<<<1024, 256, 0, stream>>>(proj_w, Wprj, 256, 256);
  pack_bias<<<1, 256, 0, stream>>>(proj_b, bprj, 256);
  pack_weight<<<1024, 256, 0, stream>>>(fc1_w, Wfc1, 1024, 256);
  pack_bias<<<4, 256, 0, stream>>>(fc1_b, bfc1, 1024);
  pack_weight<<<1024, 256, 0, stream>>>(fc2_w, Wfc2, 256, 1024);
  pack_bias<<<1, 256, 0, stream>>>(fc2_b, bfc2, 256);

  // 1) LN1 -> packed [re|im] (window partition is a no-op view)
  cln_kernel<<<TT / 8, 256, 0, stream>>>(x_r, x_i, 256, n1g, n1b, xn);

  // 2) qkv = xn @ Wqkv + b  (M=32768, K=512, N=1536)
  gemm_wmma_f32<<<dim3(1536 / 128, TT / 128), 256, 0, stream>>>(
      xn, Wqkv, bqkv, qkv, TT, 512, 1536, 0, 768,
      nullptr, nullptr, nullptr, nullptr);

  // 3) windowed complex attention + phase softmax
  attn_kernel<<<512 * 8, 256, 0, stream>>>(qkv, rpb, aout);

  // 4) x1 = shortcut + proj(aout)  (K=512, N=512, residual = x planes)
  gemm_wmma_f32<<<dim3(512 / 128, TT / 128), 256, 0, stream>>>(
      aout, Wprj, bprj, x1, TT, 512, 512, 2, 256,
      x_r, x_i, nullptr, nullptr);

  // 5) LN2 on x1 (packed layout) -> xn (reuse)
  cln_kernel<<<TT / 8, 256, 0, stream>>>(x1, x1 + 256, 512, n2g, n2b, xn);

  // 6) h1 = cprelu(fc1(xn))  (K=512, N=2048)
  gemm_wmma_f32<<<dim3(2048 / 128, TT / 128), 256, 0, stream>>>(
      xn, Wfc1, bfc1, h1, TT, 512, 2048, 1, 1024,
      nullptr, nullptr, slopes, nullptr);

  // 7) out = x1 + fc2(h1), written as stacked real/imag planes (K=2048, N=512)
  float* outR = (float*)d_out;
  float* outI = outR + (size_t)TT * 256;
  gemm_wmma_f32<<<dim3(512 / 128, TT / 128), 256, 0, stream>>>(
      h1, Wfc2, bfc2, outR, TT, 2048, 512, 3, 256,
      x1, nullptr, nullptr, outI);
}